// StrokeField_2817498546520
// MI455X (gfx1250) — compile-verified
//
#include <hip/hip_runtime.h>
#include <math.h>
#include <stdint.h>

typedef float v2f __attribute__((ext_vector_type(2)));
typedef float v4f __attribute__((ext_vector_type(4)));
typedef float v8f __attribute__((ext_vector_type(8)));

#define N_POINTS        2097152
#define N_STROKES       500
#define S_PAD           512         // strokes padded to multiple of 16
#define CHUNKS          (S_PAD / 16)
#define WAVES_PER_BLOCK 8
#define GROUPS_PER_WAVE 8
#define POINTS_PER_BLOCK (WAVES_PER_BLOCK * GROUPS_PER_WAVE * 16)  // 1024

// async copy of one 16B chunk: global address saddr + voff to the LDS address in ldsaddr
__device__ __forceinline__ void async_b128(uint32_t ldsaddr, uint32_t voff, const void* saddr) {
    asm volatile("global_load_async_to_lds_b128 %0, %1, %2"
                 :: "v"(ldsaddr), "v"(voff), "s"(saddr) : "memory");
}

__global__ __launch_bounds__(256)
void StrokeField_kernel(const float* __restrict__ coords,   // [N,3]
                        const float* __restrict__ shape,    // [S,4] cx,cy,cz,r
                        const float* __restrict__ color,    // [S,3]
                        const float* __restrict__ alpha,    // [S,1]
                        float* __restrict__ out)            // [N] + [N,3] + [N,3]
{
    // LDS-resident stroke tables (24 KB) + raw staging (16 KB) of 320 KB WGP LDS
    __shared__ float sA[S_PAD][4];   // -2cx, -2cy, -2cz, ||center||^2 (WMMA A rows)
    __shared__ float sK[S_PAD][8];   // 5r+0.5, density, cr, cg, cb, pad
    __shared__ __align__(16) float stS[N_STROKES * 4];  // raw shape
    __shared__ __align__(16) float stC[N_STROKES * 3];  // raw color
    __shared__ __align__(16) float stL[N_STROKES];      // raw alpha

    const int tid = threadIdx.x;

    // ---- async global->LDS staging of all stroke params (ASYNCcnt path) ----
    {
        const uint32_t aS = (uint32_t)(uintptr_t)&stS[0];
        const uint32_t aC = (uint32_t)(uintptr_t)&stC[0];
        const uint32_t aL = (uint32_t)(uintptr_t)&stL[0];
        #pragma unroll
        for (int i = tid; i < 500; i += 256)            // 500 x 16B = shape
            async_b128(aS + i * 16u, (uint32_t)(i * 16), (const void*)shape);
        if (tid < 119)                                  // 375 x 16B = color
            async_b128(aC + (tid + 256) * 16u, (uint32_t)((tid + 256) * 16), (const void*)color);
        if (tid < 256)
            async_b128(aC + tid * 16u, (uint32_t)(tid * 16), (const void*)color);
        if (tid < 125)                                  // 125 x 16B = alpha
            async_b128(aL + tid * 16u, (uint32_t)(tid * 16), (const void*)alpha);
        asm volatile("s_wait_asynccnt 0x0" ::: "memory");
    }
    __syncthreads();

    // ---- preprocess strokes from LDS staging into WMMA-ready tables ----
    for (int s = tid; s < S_PAD; s += 256) {
        float cx = 0.f, cy = 0.f, cz = 0.f, r = -1e30f;
        float dp = 0.f, cr = 0.f, cg = 0.f, cb = 0.f;
        if (s < N_STROKES) {
            cx = stS[s * 4 + 0]; cy = stS[s * 4 + 1];
            cz = stS[s * 4 + 2]; r  = stS[s * 4 + 3];
            cr = stC[s * 3 + 0]; cg = stC[s * 3 + 1]; cb = stC[s * 3 + 2];
            dp = fmaxf(stL[s], 0.0f) * 50.0f;           // MAX_DENSITY
        }
        sA[s][0] = -2.0f * cx; sA[s][1] = -2.0f * cy; sA[s][2] = -2.0f * cz;
        sA[s][3] = cx * cx + cy * cy + cz * cz;
        sK[s][0] = fmaf(5.0f, r, 0.5f);                 // pre-folded t intercept
        sK[s][1] = dp;
        sK[s][2] = cr; sK[s][3] = cg; sK[s][4] = cb;
        sK[s][5] = 0.f; sK[s][6] = 0.f; sK[s][7] = 0.f;
    }
    __syncthreads();

    const int  lane = tid & 31;
    const int  wave = tid >> 5;
    const bool hi   = (lane >= 16);
    const int  pl   = lane & 15;
    const int  asel = hi ? 1 : 0;   // which v2f half of the sA row
    const int  ksel = hi ? 8 : 0;   // stroke sub-range within chunk

    for (int g = 0; g < GROUPS_PER_WAVE; ++g) {
        const int pbase = ((blockIdx.x * WAVES_PER_BLOCK + wave) * GROUPS_PER_WAVE + g) * 16;
        const int p     = pbase + pl;

        if (g + 1 < GROUPS_PER_WAVE)                    // global_prefetch_b8
            __builtin_prefetch(&coords[(p + 16) * 3], 0, 0);

        // ---- warp: mip-NeRF-360 contraction then /2 (branch-free; fast sqrt/rcp
        //      are 1-ULP class, fine at unit scale vs the 0.1 SDF band)
        float x = coords[p * 3 + 0], y = coords[p * 3 + 1], z = coords[p * 3 + 2];
        float n  = fmaxf(__builtin_amdgcn_sqrtf(fmaf(x, x, fmaf(y, y, z * z))), 1e-9f);
        float rn = __builtin_amdgcn_rcpf(n);
        float sc = (n > 1.0f) ? ((2.0f - rn) * rn) : 1.0f;
        x = x * sc * 0.5f; y = y * sc * 0.5f; z = z * sc * 0.5f;
        const float n2p = fmaf(x, x, fmaf(y, y, z * z));    // ||c||^2

        // WMMA B fragment: column = (cx, cy, cz, 1) for point pl
        v2f b;
        b.x = hi ? z : x;
        b.y = hi ? 1.0f : y;

        float hd = 0.0f, hr = 0.0f, hg = 0.0f, hb = 0.0f, w = 1.0f;

        #pragma unroll 2
        for (int k = 0; k < CHUNKS; ++k) {
            // WMMA A fragment: one ds_load_b64 of the half-row this lane needs
            const int srow = k * 16 + pl;
            const v2f a = ((const v2f*)&sA[srow][0])[asel];

            v8f czero = {};
            // D[stroke][point] = ||s||^2 - 2 c.s   (16 strokes x 16 points)
            v8f d = __builtin_amdgcn_wmma_f32_16x16x4_f32(
                false, a, false, b, (short)0, czero, false, false);

            // lanes<16: strokes k*16+0..7 of point pl; lanes>=16: strokes k*16+8..15.
            // Fold 8 strokes (in order) into a local affine map h -> A*h + B.
            float A = 1.0f, Bd = 0.0f, Br = 0.0f, Bg = 0.0f, Bb = 0.0f;
            const float* sk = &sK[k * 16 + ksel][0];
            #pragma unroll
            for (int j = 0; j < 8; ++j) {
                const v4f  q    = *(const v4f*)&sk[j * 8];      // r5, dp, cr, cg
                const float cb  = sk[j * 8 + 4];
                const float d2  = fmaxf(n2p + d[j], 0.0f);
                const float dist = __builtin_amdgcn_sqrtf(d2); // raw v_sqrt_f32
                // t = clamp(5*(r-dist)+0.5) = fma(-5, dist, r5) with clamp modifier
                const float t   = __builtin_amdgcn_fmed3f(fmaf(-5.0f, dist, q.x), 0.0f, 1.0f);
                const float omt = 1.0f - t;
                A  = A * omt;
                Bd = fmaf(omt, Bd, t * q.y);
                Br = fmaf(omt, Br, t * q.z);
                Bg = fmaf(omt, Bg, t * q.w);
                Bb = fmaf(omt, Bb, t * cb);
            }

            // Exchange with partner half-wave and compose lo-then-hi (stroke order).
            const float pA  = __shfl_xor(A, 16, 32);
            const float pBd = __shfl_xor(Bd, 16, 32);
            const float pBr = __shfl_xor(Br, 16, 32);
            const float pBg = __shfl_xor(Bg, 16, 32);
            const float pBb = __shfl_xor(Bb, 16, 32);
            const float loA  = hi ? pA  : A,  hiA  = hi ? A  : pA;
            const float loBd = hi ? pBd : Bd, hiBd = hi ? Bd : pBd;
            const float loBr = hi ? pBr : Br, hiBr = hi ? Br : pBr;
            const float loBg = hi ? pBg : Bg, hiBg = hi ? Bg : pBg;
            const float loBb = hi ? pBb : Bb, hiBb = hi ? Bb : pBb;
            const float cA  = hiA * loA;
            const float cBd = fmaf(hiA, loBd, hiBd);
            const float cBr = fmaf(hiA, loBr, hiBr);
            const float cBg = fmaf(hiA, loBg, hiBg);
            const float cBb = fmaf(hiA, loBb, hiBb);

            hd = fmaf(cA, hd, cBd);
            hr = fmaf(cA, hr, cBr);
            hg = fmaf(cA, hg, cBg);
            hb = fmaf(cA, hb, cBb);
            w  = w * cA;
        }

        // rgb = clip(h_color / (1 + 1e-6 - w), 0, 1)
        const float inv = __builtin_amdgcn_rcpf(1.0f + 1e-6f - w);
        const float rr  = __builtin_amdgcn_fmed3f(hr * inv, 0.0f, 1.0f);
        const float rg  = __builtin_amdgcn_fmed3f(hg * inv, 0.0f, 1.0f);
        const float rb  = __builtin_amdgcn_fmed3f(hb * inv, 0.0f, 1.0f);

        // both halves hold identical state; split the stores across the halves
        if (!hi) {
            out[p] = hd;
            out[N_POINTS + p * 3 + 0] = rr;
            out[N_POINTS + p * 3 + 1] = rg;
            out[N_POINTS + p * 3 + 2] = rb;
        } else {
            out[4 * N_POINTS + p * 3 + 0] = x;
            out[4 * N_POINTS + p * 3 + 1] = y;
            out[4 * N_POINTS + p * 3 + 2] = z;
        }
    }
}

extern "C" void kernel_launch(void* const* d_in, const int* in_sizes, int n_in,
                              void* d_out, int out_size, void* d_ws, size_t ws_size,
                              hipStream_t stream) {
    const float* coords = (const float*)d_in[0];
    const float* shape  = (const float*)d_in[1];
    const float* color  = (const float*)d_in[2];
    const float* alpha  = (const float*)d_in[3];
    float* out = (float*)d_out;
    (void)in_sizes; (void)n_in; (void)out_size; (void)d_ws; (void)ws_size;

    const int blocks = N_POINTS / POINTS_PER_BLOCK;   // 2048, exact cover
    StrokeField_kernel<<<blocks, 256, 0, stream>>>(coords, shape, color, alpha, out);
}